// Agent_68959994905374
// MI455X (gfx1250) — compile-verified
//
#include <hip/hip_runtime.h>
#include <math.h>

typedef __attribute__((ext_vector_type(2))) float v2f;
typedef __attribute__((ext_vector_type(8))) float v8f;
typedef __attribute__((ext_vector_type(4))) unsigned int v4u;
typedef __attribute__((ext_vector_type(4))) int v4i;
typedef __attribute__((ext_vector_type(8))) int v8i;

#ifndef __has_builtin
#define __has_builtin(x) 0
#endif
#if defined(__gfx1250__) && __has_builtin(__builtin_amdgcn_tensor_load_to_lds)
#define USE_TDM 1
#else
#define USE_TDM 0
#endif

constexpr int   kB   = 128;
constexpr int   kOBS = 128;
constexpr int   kH   = 64;
constexpr int   kNR  = 102;
constexpr int   kV   = kNR * kNR;   // 10404
constexpr int   kA   = kV * 3;      // 31212
constexpr float kLogSqrt2Pi = 0.91893853320467274178f;

// ---------- helpers ----------
__device__ __forceinline__ float block_reduce_256(float v) {
    __shared__ float red[8];
    int lane = threadIdx.x & 31;
    int w    = threadIdx.x >> 5;
    if (threadIdx.x < 8) red[threadIdx.x] = 0.0f;
    __syncthreads();
#pragma unroll
    for (int off = 16; off >= 1; off >>= 1) v += __shfl_xor(v, off, 32);
    if (lane == 0) red[w] = v;
    __syncthreads();
    float s = 0.0f;
    if (threadIdx.x < 8) s = red[threadIdx.x];
#pragma unroll
    for (int off = 4; off >= 1; off >>= 1) s += __shfl_xor(s, off, 32);
    return s;   // valid on thread 0
}

// ---------- zero scratch ----------
__global__ void k_zero(float* p, long n) {
    long i = (long)blockIdx.x * blockDim.x + threadIdx.x;
    if (i < n) p[i] = 0.0f;
}

// ---------- small MLPs (actor hidden + critic value), one block per batch row ----------
__global__ void k_mlp(const float* __restrict__ x,
                      const float* __restrict__ aw1, const float* __restrict__ ab1,
                      const float* __restrict__ aw2, const float* __restrict__ ab2,
                      const float* __restrict__ cw1, const float* __restrict__ cb1,
                      const float* __restrict__ cw2, const float* __restrict__ cb2,
                      const float* __restrict__ cw3, const float* __restrict__ cb3,
                      float* __restrict__ h_out, float* __restrict__ value_out) {
    int b = blockIdx.x;
    int j = threadIdx.x;                 // 0..63
    __shared__ float xrow[kOBS];
    __shared__ float h1a[kH], h1c[kH], val[kH];
    for (int i = j; i < kOBS; i += kH) xrow[i] = x[b * kOBS + i];
    __syncthreads();

    float sa = ab1[j], sc = cb1[j];
    for (int k = 0; k < kOBS; ++k) {
        float xv = xrow[k];
        sa += xv * aw1[k * kH + j];
        sc += xv * cw1[k * kH + j];
    }
    h1a[j] = tanhf(sa);
    h1c[j] = tanhf(sc);
    __syncthreads();

    float s2a = ab2[j], s2c = cb2[j];
    for (int k = 0; k < kH; ++k) {
        s2a += h1a[k] * aw2[k * kH + j];
        s2c += h1c[k] * cw2[k * kH + j];
    }
    float h2a = tanhf(s2a);
    float h2c = tanhf(s2c);
    h_out[b * kH + j] = h2a;
    val[j] = h2c * cw3[j];
    __syncthreads();
    if (j == 0) {
        float s = cb3[0];
        for (int k = 0; k < kH; ++k) s += val[k];
        value_out[b] = s;
    }
}

// ---------- sum(logstd) ----------
__global__ void k_logstd_sum(const float* __restrict__ logstd, float* __restrict__ S) {
    int i = blockIdx.x * blockDim.x + threadIdx.x;
    float v = (i < kA) ? logstd[i] : 0.0f;
    float s = block_reduce_256(v);
    if (threadIdx.x == 0) atomicAdd(S, s);
}

// ---------- fused WMMA GEMM (mean) + log_prob partials + action copy + verts ----------
// block = 256 threads (8 waves); each wave owns a 16x16 tile: full M=128 strip x 16 cols.
// The 64x16 B-strip of w3 is staged into LDS once per block by the Tensor Data Mover
// (tensor_load_to_lds, TENSORcnt), removing the 8x per-wave redundant global reads.
__global__ void k_gemm_logprob(const float* __restrict__ h,     // [128][64]
                               const float* __restrict__ w3,    // [64][A]
                               const float* __restrict__ b3,    // [A]
                               const float* __restrict__ action,// [128*A]
                               const float* __restrict__ logstd,// [A]
                               const float* __restrict__ vertices, // [128*A]
                               float* __restrict__ out_action,
                               float* __restrict__ verts,
                               float* __restrict__ Q) {
    const int wave   = threadIdx.x >> 5;      // 0..7  -> m_base
    const int lane   = threadIdx.x & 31;
    const int l16    = lane >> 4;             // 0: lanes 0-15, 1: lanes 16-31
    const int m_base = wave * 16;
    const int n_base = blockIdx.x * 16;
    const int mrow   = m_base + (lane & 15);  // A-frag row
    const int ncol   = n_base + (lane & 15);  // B/D-frag column
    const int ncl    = (ncol < kA) ? ncol : (kA - 1);

#if USE_TDM
    __shared__ float lds_b[kH * 16];          // 64 rows x 16 cols = 4KB
    if (wave == 0) {
        if (threadIdx.x == 0) {
            // ---- Tensor DMA descriptor (D#), ISA ch.8 bit layout ----
            unsigned long long gaddr = (unsigned long long)(const void*)(w3 + n_base);
            unsigned lds_addr = (unsigned)(unsigned long long)(void*)lds_b; // flat LDS addr[31:0] == LDS byte offset
            unsigned td0 = (unsigned)(kA - n_base);  // remaining width: OOB columns zero-fill
            v4u g0 = { 1u,                                             // count=1
                       lds_addr,                                       // [63:32] lds_addr
                       (unsigned)(gaddr & 0xffffffffu),                // global_addr[31:0]
                       (unsigned)((gaddr >> 32) & 0x01ffffffu) | (2u << 30) }; // addr[56:32] | type=2
            v8i g1 = { (int)0x00020000u,                               // data_size=2 (4B)
                       (int)((td0 & 0xffffu) << 16),                   // tensor_dim0[15:0]
                       (int)((td0 >> 16) | (64u << 16)),               // tensor_dim0[31:16] | tensor_dim1 lo (=64)
                       (int)(16u << 16),                               // tensor_dim1 hi=0 | tile_dim0=16
                       (int)64u,                                       // tile_dim1=64 | tile_dim2=0
                       (int)kA,                                        // tensor_dim0_stride[31:0]
                       0, 0 };                                         // stride hi / dim1_stride unused (2D)
            v4i gz = { 0, 0, 0, 0 };
#if __clang_major__ >= 23
            v8i gz8 = { 0, 0, 0, 0, 0, 0, 0, 0 };
            __builtin_amdgcn_tensor_load_to_lds(g0, g1, gz, gz, gz8, 0);
#else
            __builtin_amdgcn_tensor_load_to_lds(g0, g1, gz, gz, 0);
#endif
        }
        __builtin_amdgcn_s_wait_tensorcnt(0);
    }
    __syncthreads();
#endif

    v8f c = {};
#pragma unroll
    for (int k0 = 0; k0 < kH; k0 += 4) {
        const int ka = k0 + l16 * 2;
        v2f a, bb;
        a.x  = h[mrow * kH + ka];
        a.y  = h[mrow * kH + ka + 1];
#if USE_TDM
        bb.x = lds_b[ka * 16 + (lane & 15)];
        bb.y = lds_b[(ka + 1) * 16 + (lane & 15)];
#else
        bb.x = w3[(size_t)ka * kA + ncl];
        bb.y = w3[(size_t)(ka + 1) * kA + ncl];
#endif
        c = __builtin_amdgcn_wmma_f32_16x16x4_f32(false, a, false, bb,
                                                  (short)0, c, false, false);
    }

    const float bias    = b3[ncl];
    const float inv_std = __expf(-logstd[ncl]);
#pragma unroll
    for (int r = 0; r < 8; ++r) {
        const int row = m_base + r + l16 * 8;   // D layout: VGPR r -> M=r (+8 upper lanes)
        float q = 0.0f;
        if (ncol < kA) {
            const size_t idx = (size_t)row * kA + ncol;
            const float mean = c[r] + bias;
            const float act  = action[idx];
            out_action[idx]  = act;                    // pass-through output
            verts[idx]       = vertices[idx] + act;    // fused verts
            const float t    = (act - mean) * inv_std;
            q = t * t;
        }
#pragma unroll
        for (int off = 8; off >= 1; off >>= 1) q += __shfl_xor(q, off, 16);
        if ((lane & 15) == 0) atomicAdd(&Q[row], q);
    }
}

// ---------- vertex degree ----------
__global__ void k_deg(const int* __restrict__ edges, int nE, float* __restrict__ deg) {
    int e = blockIdx.x * blockDim.x + threadIdx.x;
    if (e < nE) {
        atomicAdd(&deg[edges[2 * e + 0]], 1.0f);
        atomicAdd(&deg[edges[2 * e + 1]], 1.0f);
    }
}

// ---------- edge loss + neighbor scatter ----------
__global__ void k_edge(const float* __restrict__ verts, const int* __restrict__ edges,
                       int nE, float* __restrict__ nbr, float* __restrict__ loss_edge) {
    long t = (long)blockIdx.x * blockDim.x + threadIdx.x;
    float d2 = 0.0f;
    if (t < (long)kB * nE) {
        int b = (int)(t / nE);
        int e = (int)(t - (long)b * nE);
        int e0 = edges[2 * e + 0], e1 = edges[2 * e + 1];
        const float* p0 = verts + ((size_t)b * kV + e0) * 3;
        const float* p1 = verts + ((size_t)b * kV + e1) * 3;
        float dx = p0[0] - p1[0], dy = p0[1] - p1[1], dz = p0[2] - p1[2];
        d2 = dx * dx + dy * dy + dz * dz;
        float* n0 = nbr + ((size_t)b * kV + e0) * 3;
        float* n1 = nbr + ((size_t)b * kV + e1) * 3;
        atomicAdd(&n0[0], p1[0]); atomicAdd(&n0[1], p1[1]); atomicAdd(&n0[2], p1[2]);
        atomicAdd(&n1[0], p0[0]); atomicAdd(&n1[1], p0[1]); atomicAdd(&n1[2], p0[2]);
    }
    float s = block_reduce_256(d2);
    if (threadIdx.x == 0) atomicAdd(loss_edge, s);
}

// ---------- dihedral/normal loss ----------
__global__ void k_normal(const float* __restrict__ verts, const int* __restrict__ interior,
                         int nI, float* __restrict__ loss_normal) {
    long t = (long)blockIdx.x * blockDim.x + threadIdx.x;
    float contrib = 0.0f;
    if (t < (long)kB * nI) {
        int b = (int)(t / nI);
        int i = (int)(t - (long)b * nI);
        int i0 = interior[4 * i + 0], i1 = interior[4 * i + 1];
        int ia = interior[4 * i + 2], ib = interior[4 * i + 3];
        const float* v0 = verts + ((size_t)b * kV + i0) * 3;
        const float* v1 = verts + ((size_t)b * kV + i1) * 3;
        const float* pa = verts + ((size_t)b * kV + ia) * 3;
        const float* pb = verts + ((size_t)b * kV + ib) * 3;
        float ux = v1[0] - v0[0], uy = v1[1] - v0[1], uz = v1[2] - v0[2];
        float ax = pa[0] - v0[0], ay = pa[1] - v0[1], az = pa[2] - v0[2];
        float bx = pb[0] - v0[0], by = pb[1] - v0[1], bz = pb[2] - v0[2];
        // n0 = cross(u, a); n1 = cross(b, u)
        float n0x = uy * az - uz * ay, n0y = uz * ax - ux * az, n0z = ux * ay - uy * ax;
        float n1x = by * uz - bz * uy, n1y = bz * ux - bx * uz, n1z = bx * uy - by * ux;
        float dot = n0x * n1x + n0y * n1y + n0z * n1z;
        float nn  = sqrtf(n0x * n0x + n0y * n0y + n0z * n0z) *
                    sqrtf(n1x * n1x + n1y * n1y + n1z * n1z);
        float cosv = dot / fmaxf(nn, 1e-8f);
        contrib = 1.0f - cosv;
    }
    float s = block_reduce_256(contrib);
    if (threadIdx.x == 0) atomicAdd(loss_normal, s);
}

// ---------- Laplacian loss ----------
__global__ void k_lap(const float* __restrict__ verts, const float* __restrict__ nbr,
                      const float* __restrict__ deg, float* __restrict__ loss_lap) {
    long t = (long)blockIdx.x * blockDim.x + threadIdx.x;
    float nrm = 0.0f;
    if (t < (long)kB * kV) {
        int b = (int)(t / kV);
        int v = (int)(t - (long)b * kV);
        float dg = fmaxf(deg[v], 1.0f);
        size_t base = ((size_t)b * kV + v) * 3;
        float lx = nbr[base + 0] / dg - verts[base + 0];
        float ly = nbr[base + 1] / dg - verts[base + 1];
        float lz = nbr[base + 2] / dg - verts[base + 2];
        nrm = sqrtf(lx * lx + ly * ly + lz * lz);
    }
    float s = block_reduce_256(nrm);
    if (threadIdx.x == 0) atomicAdd(loss_lap, s);
}

// ---------- finalize: log_prob, entropy, loss ----------
__global__ void k_final(const float* __restrict__ Q, const float* __restrict__ S,
                        const float* __restrict__ losses, int nE, int nI,
                        float* __restrict__ log_prob, float* __restrict__ entropy,
                        float* __restrict__ loss) {
    int b = threadIdx.x;
    float sls = S[0];
    if (b < kB) {
        log_prob[b] = -0.5f * Q[b] - sls - (float)kA * kLogSqrt2Pi;
        entropy[b]  = (float)kA * (0.5f + kLogSqrt2Pi) + sls;
    }
    if (b == 0) {
        float le = losses[0] / ((float)kB * (float)nE);
        float ln = losses[1] / ((float)kB * (float)nI);
        float ll = losses[2] / ((float)kB * (float)kV);
        loss[0] = le + ln + ll;
    }
}

extern "C" void kernel_launch(void* const* d_in, const int* in_sizes, int n_in,
                              void* d_out, int out_size, void* d_ws, size_t ws_size,
                              hipStream_t stream) {
    (void)n_in; (void)out_size; (void)ws_size;
    const float* x        = (const float*)d_in[0];
    const float* action   = (const float*)d_in[1];
    const float* vertices = (const float*)d_in[2];
    const float* cw1 = (const float*)d_in[3];
    const float* cb1 = (const float*)d_in[4];
    const float* cw2 = (const float*)d_in[5];
    const float* cb2 = (const float*)d_in[6];
    const float* cw3 = (const float*)d_in[7];
    const float* cb3 = (const float*)d_in[8];
    const float* aw1 = (const float*)d_in[9];
    const float* ab1 = (const float*)d_in[10];
    const float* aw2 = (const float*)d_in[11];
    const float* ab2 = (const float*)d_in[12];
    const float* aw3 = (const float*)d_in[13];
    const float* ab3 = (const float*)d_in[14];
    const float* logstd   = (const float*)d_in[15];
    const int*   edges    = (const int*)d_in[16];
    const int*   interior = (const int*)d_in[17];
    const int nE = in_sizes[16] / 2;
    const int nI = in_sizes[17] / 4;

    float* out        = (float*)d_out;
    float* out_action = out;
    float* out_lp     = out + (size_t)kB * kA;
    float* out_ent    = out_lp + kB;
    float* out_val    = out_ent + kB;
    float* out_loss   = out_val + kB;

    float* ws       = (float*)d_ws;
    float* ws_h     = ws;                              // B*H
    float* ws_verts = ws_h + kB * kH;                  // B*A
    float* ws_nbr   = ws_verts + (size_t)kB * kA;      // B*A
    float* ws_deg   = ws_nbr + (size_t)kB * kA;        // V
    float* ws_Q     = ws_deg + kV;                     // B
    float* ws_S     = ws_Q + kB;                       // 1
    float* ws_loss  = ws_S + 1;                        // 3

    const long nzero = (long)kB * kA + kV + kB + 1 + 3; // nbr..end contiguous
    k_zero<<<(int)((nzero + 255) / 256), 256, 0, stream>>>(ws_nbr, nzero);

    k_mlp<<<kB, kH, 0, stream>>>(x, aw1, ab1, aw2, ab2,
                                 cw1, cb1, cw2, cb2, cw3, cb3, ws_h, out_val);
    k_logstd_sum<<<(kA + 255) / 256, 256, 0, stream>>>(logstd, ws_S);
    k_gemm_logprob<<<(kA + 15) / 16, 256, 0, stream>>>(ws_h, aw3, ab3, action, logstd,
                                                       vertices, out_action, ws_verts, ws_Q);
    k_deg<<<(nE + 255) / 256, 256, 0, stream>>>(edges, nE, ws_deg);

    const long nBE = (long)kB * nE;
    k_edge<<<(int)((nBE + 255) / 256), 256, 0, stream>>>(ws_verts, edges, nE, ws_nbr, &ws_loss[0]);
    const long nBI = (long)kB * nI;
    k_normal<<<(int)((nBI + 255) / 256), 256, 0, stream>>>(ws_verts, interior, nI, &ws_loss[1]);
    const long nBV = (long)kB * kV;
    k_lap<<<(int)((nBV + 255) / 256), 256, 0, stream>>>(ws_verts, ws_nbr, ws_deg, &ws_loss[2]);

    k_final<<<1, 128, 0, stream>>>(ws_Q, ws_S, ws_loss, nE, nI, out_lp, out_ent, out_loss);
}